// DiffusionTimePredictor_62156766708185
// MI455X (gfx1250) — compile-verified
//
#include <hip/hip_runtime.h>
#include <math.h>

typedef __attribute__((ext_vector_type(2))) float v2f;
typedef __attribute__((ext_vector_type(8))) float v8f;

#define SQ 2048
#define SK 2048
#define DK 64
#define NHEADS 32
#define QCHUNKS 8
#define STRIPS_PER_WAVE 2   // 128 strips / (8 chunks * 8 waves)
#define LOG_T_MAX 2.302585092994046f

__device__ __forceinline__ float gelu_exact(float x) {
  return 0.5f * x * (1.0f + erff(x * 0.7071067811865476f));
}

// One block = (head, q-chunk). 8 waves, each wave computes full-row softmax
// stats for 2 strips of 16 query rows via WMMA f32 16x16x4 + online softmax.
__global__ __launch_bounds__(256) void head_stats_kernel(
    const float* __restrict__ query, const float* __restrict__ key,
    float* __restrict__ ws) {
  const int head  = blockIdx.x;
  const int chunk = blockIdx.y;
  const int tid   = threadIdx.x;
  const int wave  = tid >> 5;
  const int lane  = tid & 31;
  const int half  = lane >> 4;   // 0: lanes 0-15, 1: lanes 16-31
  const int m16   = lane & 15;

  const float* Qh = query + (size_t)head * SQ * DK;
  const float* Kh = key   + (size_t)head * SK * DK;

  float acc_sum = 0.f, acc_max = 0.f, acc_var = 0.f;

  const float scale  = 0.125f;                 // 1/sqrt(64)
  const float invN   = 1.0f / (float)SK;
  const float invNm1 = 1.0f / (float)(SK - 1);

  for (int sidx = 0; sidx < STRIPS_PER_WAVE; ++sidx) {
    const int strip = chunk * 16 + wave * STRIPS_PER_WAVE + sidx;
    const int q0 = strip * 16;

    // A-panel: 16 rows x 64 of Q in CDNA5 f32 16x16x4 A layout.
    // v2f a[kk]: lane (half,m16) holds Q[q0+m16][4*kk + 2*half + {0,1}]
    v2f a[16];
    const float* qbase = Qh + (size_t)(q0 + m16) * DK + 2 * half;
#pragma unroll
    for (int kk = 0; kk < 16; ++kk)
      a[kk] = *(const v2f*)(qbase + 4 * kk);

    // per-(lane,slot) online-softmax partial stats over this lane's columns
    float s_sum[8], s_max[8], s_z[8], s_s2[8];
#pragma unroll
    for (int r = 0; r < 8; ++r) {
      s_sum[r] = 0.f; s_max[r] = -INFINITY; s_z[r] = 0.f; s_s2[r] = 0.f;
    }

    const float* kbase = Kh + (size_t)m16 * DK + 2 * half;
    for (int tile = 0; tile < SK / 16; ++tile) {
      const float* kb = kbase + (size_t)tile * 16 * DK;
      // B-panel: K^T tile, same per-lane layout (B[k][n] = Key[n][k])
      v2f b[16];
#pragma unroll
      for (int kk = 0; kk < 16; ++kk)
        b[kk] = *(const v2f*)(kb + 4 * kk);

      v8f c = {};
#pragma unroll
      for (int kk = 0; kk < 16; ++kk)
        c = __builtin_amdgcn_wmma_f32_16x16x4_f32(false, a[kk], false, b[kk],
                                                  (short)0, c, false, false);

      // c[r]: row q0 + r + 8*half, column tile*16 + m16
#pragma unroll
      for (int r = 0; r < 8; ++r) {
        float s  = c[r] * scale;
        float mn = fmaxf(s_max[r], s);
        float eo = __expf(s_max[r] - mn);   // rescale old terms
        float en = __expf(s - mn);
        s_z[r]  = s_z[r]  * eo + en;
        s_s2[r] = s_s2[r] * eo * eo + en * en;
        s_sum[r] += s;
        s_max[r] = mn;
      }
    }

    // merge the 16 per-column partials across each 16-lane half
#pragma unroll
    for (int off = 8; off > 0; off >>= 1) {
#pragma unroll
      for (int r = 0; r < 8; ++r) {
        float mo = __shfl_xor(s_max[r], off, 16);
        float zo = __shfl_xor(s_z[r],   off, 16);
        float so = __shfl_xor(s_s2[r],  off, 16);
        float uo = __shfl_xor(s_sum[r], off, 16);
        float mn = fmaxf(s_max[r], mo);
        float e0 = __expf(s_max[r] - mn);
        float e1 = __expf(mo - mn);
        s_z[r]  = s_z[r]  * e0 + zo * e1;
        s_s2[r] = s_s2[r] * e0 * e0 + so * e1 * e1;
        s_sum[r] += uo;
        s_max[r] = mn;
      }
    }

    // every lane in a half now holds full-row stats for its 8 rows
#pragma unroll
    for (int r = 0; r < 8; ++r) {
      acc_sum += s_sum[r];
      acc_max += s_max[r];
      acc_var += (s_s2[r] / (s_z[r] * s_z[r]) - invN) * invNm1;
    }
  }

  // deterministic block reduction: 16 (wave,half) partials summed in order
  __shared__ float parts[16][3];
  if (m16 == 0) {
    parts[wave * 2 + half][0] = acc_sum;
    parts[wave * 2 + half][1] = acc_max;
    parts[wave * 2 + half][2] = acc_var;
  }
  __syncthreads();
  if (tid == 0) {
    float t0 = 0.f, t1 = 0.f, t2 = 0.f;
    for (int i = 0; i < 16; ++i) {
      t0 += parts[i][0]; t1 += parts[i][1]; t2 += parts[i][2];
    }
    float* w = ws + ((size_t)head * QCHUNKS + chunk) * 3;
    w[0] = t0; w[1] = t1; w[2] = t2;
  }
}

// One block per head: fold chunk partials -> features -> 3->64->64->1 MLP.
__global__ __launch_bounds__(64) void mlp_kernel(
    const float* __restrict__ ws,
    const float* __restrict__ W1, const float* __restrict__ b1,
    const float* __restrict__ W2, const float* __restrict__ b2,
    const float* __restrict__ W3, const float* __restrict__ b3,
    float* __restrict__ out) {
  const int head = blockIdx.x;
  const int j = threadIdx.x;
  __shared__ float feat[3], h1[64], h2[64];

  if (j < 3) {
    float s = 0.f;
    for (int c = 0; c < QCHUNKS; ++c)
      s += ws[((size_t)head * QCHUNKS + c) * 3 + j];
    s *= (j == 0) ? (1.0f / ((float)SQ * (float)SK)) : (1.0f / (float)SQ);
    feat[j] = s;
  }
  __syncthreads();

  float x = b1[j];
  for (int i = 0; i < 3; ++i) x += feat[i] * W1[i * 64 + j];
  h1[j] = gelu_exact(x);
  __syncthreads();

  float y = b2[j];
  for (int i = 0; i < 64; ++i) y += h1[i] * W2[i * 64 + j];
  h2[j] = gelu_exact(y);
  __syncthreads();

  if (j == 0) {
    float t = b3[0];
    for (int i = 0; i < 64; ++i) t += h2[i] * W3[i];
    t = fminf(fmaxf(t, -LOG_T_MAX), LOG_T_MAX);
    out[head] = __expf(t);
  }
}

extern "C" void kernel_launch(void* const* d_in, const int* in_sizes, int n_in,
                              void* d_out, int out_size, void* d_ws, size_t ws_size,
                              hipStream_t stream) {
  (void)in_sizes; (void)n_in; (void)out_size; (void)ws_size;
  const float* query = (const float*)d_in[0];
  const float* key   = (const float*)d_in[1];
  const float* W1    = (const float*)d_in[2];
  const float* b1    = (const float*)d_in[3];
  const float* W2    = (const float*)d_in[4];
  const float* b2    = (const float*)d_in[5];
  const float* W3    = (const float*)d_in[6];
  const float* b3    = (const float*)d_in[7];
  float* out = (float*)d_out;
  float* ws  = (float*)d_ws;   // NHEADS * QCHUNKS * 3 floats = 3 KB

  head_stats_kernel<<<dim3(NHEADS, QCHUNKS), 256, 0, stream>>>(query, key, ws);
  mlp_kernel<<<NHEADS, 64, 0, stream>>>(ws, W1, b1, W2, b2, W3, b3, out);
}